// GraphConv_2774548873916
// MI455X (gfx1250) — compile-verified
//
#include <hip/hip_runtime.h>

// ---------------------------------------------------------------------------
// GraphSAGE forward for MI455X (gfx1250, wave32).
// Dense matmuls: V_WMMA_F32_16X16X4_F32 (f32 kept end-to-end; compute is not
// the bottleneck). Weights staged through LDS (ds_load) once per block.
// Edge aggregation: no-return global_atomic_add_f32 (L2-side atomics).
// ---------------------------------------------------------------------------

typedef __attribute__((ext_vector_type(2))) float v2f;
typedef __attribute__((ext_vector_type(8))) float v8f;

#define N_NODES  50000
#define N_EDGES  800000
#define N_GRAPHS 1000
#define NEG_SLOPE 0.01f
#define WAVES_PER_BLOCK 4
#define KMAX 128            // max K of any layer (emb / layer1 / final)

// ---------------------------------------------------------------------------
// Fused GEMM:  out[M,N] = A1[M,K1] @ W1[K1,N] (+ A2[M,K2] @ W2[K2,N]) + bias
//
// Block = 4 waves sharing one column tile (tileN), covering 4 adjacent row
// tiles. The K1x16 / K2x16 weight slabs are staged into LDS once per block;
// each wave then issues one 8-byte global A load + two ds_load_b32 B loads
// per V_WMMA_F32_16X16X4_F32.
//
// f32 WMMA operand layout (ISA 7.12.2):
//   A (16x4):  lanes 0-15 -> M=lane,    VGPRs = K(k0,k0+1); lanes16-31 -> K+2
//   B (4x16):  lanes 0-15 -> N=lane,    VGPRs = K rows (k0,k0+1); hi lanes +2
//   C/D:       VGPR r: lanes 0-15 = (M=r, N=lane); lanes16-31 = (M=8+r, ...)
//
// GUARD=true adds branch-free row clamping/masking (EXEC stays all-ones
// around every WMMA); used only for the M=1000 final GEMM.
// ---------------------------------------------------------------------------
template <bool GUARD>
__global__ __launch_bounds__(32 * WAVES_PER_BLOCK)
void fused_gemm_wmma_kernel(const float* __restrict__ A1,
                            const float* __restrict__ W1,
                            const float* __restrict__ A2,
                            const float* __restrict__ W2,
                            const float* __restrict__ bias,
                            float* __restrict__ out,
                            int M, int K1, int K2, int N, int leaky)
{
    __shared__ float lB1[KMAX * 16];
    __shared__ float lB2[KMAX * 16];

    const int tid   = threadIdx.x;
    const int lane  = tid & 31;
    const int wave  = tid >> 5;
    const int tileN = blockIdx.x;                         // N / 16 tiles
    const int tileM = blockIdx.y * WAVES_PER_BLOCK + wave;
    const int rowTiles = (M + 15) >> 4;
    const int colBase  = tileN * 16;

    // ---- stage weight slabs (columns colBase..colBase+15) into LDS --------
    for (int i = tid; i < K1 * 16; i += 32 * WAVES_PER_BLOCK) {
        lB1[i] = W1[(size_t)(i >> 4) * N + colBase + (i & 15)];
    }
    if (A2 != nullptr) {
        for (int i = tid; i < K2 * 16; i += 32 * WAVES_PER_BLOCK) {
            lB2[i] = W2[(size_t)(i >> 4) * N + colBase + (i & 15)];
        }
    }
    __syncthreads();

    if (tileM >= rowTiles) return;                        // uniform per wave

    const int halfSel = lane >> 4;                        // 0 | 1
    const int l15     = lane & 15;
    const int col     = colBase + l15;                    // N index this lane owns
    const int rowA    = tileM * 16 + l15;                 // A row this lane reads

    int   rowAc  = rowA;
    float rowMsk = 1.0f;
    if (GUARD) {
        rowAc  = rowA < M ? rowA : (M - 1);
        rowMsk = rowA < M ? 1.0f : 0.0f;
    }

    v8f acc = {};

#pragma unroll 4
    for (int k0 = 0; k0 < K1; k0 += 4) {
        const int kb = k0 + halfSel * 2;
        v2f a = *(const v2f*)(A1 + (size_t)rowAc * K1 + kb);   // 8B global load
        if (GUARD) { a.x *= rowMsk; a.y *= rowMsk; }
        v2f b;
        b.x = lB1[kb * 16 + l15];                              // ds_load_b32
        b.y = lB1[(kb + 1) * 16 + l15];
        acc = __builtin_amdgcn_wmma_f32_16x16x4_f32(
                  false, a, false, b, (short)0, acc, false, false);
    }
    if (A2 != nullptr) {
#pragma unroll 4
        for (int k0 = 0; k0 < K2; k0 += 4) {
            const int kb = k0 + halfSel * 2;
            v2f a = *(const v2f*)(A2 + (size_t)rowAc * K2 + kb);
            if (GUARD) { a.x *= rowMsk; a.y *= rowMsk; }
            v2f b;
            b.x = lB2[kb * 16 + l15];
            b.y = lB2[(kb + 1) * 16 + l15];
            acc = __builtin_amdgcn_wmma_f32_16x16x4_f32(
                      false, a, false, b, (short)0, acc, false, false);
        }
    }

    const float bv = bias ? bias[col] : 0.0f;
#pragma unroll
    for (int r = 0; r < 8; ++r) {
        const int m = tileM * 16 + halfSel * 8 + r;
        float v = acc[r] + bv;
        if (leaky) v = (v >= 0.0f) ? v : NEG_SLOPE * v;
        if (!GUARD || m < M) out[(size_t)m * N + col] = v;
    }
}

// ---------------------------------------------------------------------------
// agg[dst[e], :] += h[src[e], :]   (segment_sum over edges)
// One thread per (edge, 4-channel chunk): float4 gather + 4 f32 atomics.
// ---------------------------------------------------------------------------
__global__ void scatter_add_kernel(const float* __restrict__ h,
                                   const int* __restrict__ src,
                                   const int* __restrict__ dst,
                                   float* __restrict__ agg,
                                   int nEdges, int d4, int d)
{
    long long idx   = (long long)blockIdx.x * blockDim.x + threadIdx.x;
    long long total = (long long)nEdges * d4;
    if (idx >= total) return;
    const int e = (int)(idx / d4);
    const int c = (int)(idx % d4) * 4;
    const int s = src[e];
    const int t = dst[e];
    const float4 v = *(const float4*)(h + (size_t)s * d + c);
    float* q = agg + (size_t)t * d + c;
    atomicAdd(q + 0, v.x);
    atomicAdd(q + 1, v.y);
    atomicAdd(q + 2, v.z);
    atomicAdd(q + 3, v.w);
}

__global__ void zero_kernel(float* __restrict__ p, long long n)
{
    long long i = (long long)blockIdx.x * blockDim.x + threadIdx.x;
    if (i < n) p[i] = 0.0f;
}

// ---------------------------------------------------------------------------
// Pooling: sums[batch[i], :] += h[i, :];  cnts[batch[i]] += 1
// ---------------------------------------------------------------------------
__global__ void pool_sum_kernel(const float* __restrict__ h,
                                const int* __restrict__ batch,
                                float* __restrict__ sums,
                                float* __restrict__ cnts,
                                int nNodes, int d4, int d)
{
    long long idx   = (long long)blockIdx.x * blockDim.x + threadIdx.x;
    long long total = (long long)nNodes * d4;
    if (idx >= total) return;
    const int node = (int)(idx / d4);
    const int c    = (int)(idx % d4) * 4;
    const int g    = batch[node];
    const float4 v = *(const float4*)(h + (size_t)node * d + c);
    float* q = sums + (size_t)g * d + c;
    atomicAdd(q + 0, v.x);
    atomicAdd(q + 1, v.y);
    atomicAdd(q + 2, v.z);
    atomicAdd(q + 3, v.w);
    if (c == 0) atomicAdd(&cnts[g], 1.0f);
}

__global__ void pool_div_kernel(const float* __restrict__ sums,
                                const float* __restrict__ cnts,
                                float* __restrict__ pooled,
                                int G, int d)
{
    int i = blockIdx.x * blockDim.x + threadIdx.x;
    if (i >= G * d) return;
    const int g = i / d;
    const float c = cnts[g];
    pooled[i] = sums[i] / (c > 1.0f ? c : 1.0f);
}

// ---------------------------------------------------------------------------
// Host-side orchestration
// ---------------------------------------------------------------------------
static inline dim3 gemm_grid(int M, int N)
{
    const int rowTiles = (M + 15) / 16;
    return dim3((unsigned)(N / 16),
                (unsigned)((rowTiles + WAVES_PER_BLOCK - 1) / WAVES_PER_BLOCK));
}

extern "C" void kernel_launch(void* const* d_in, const int* in_sizes, int n_in,
                              void* d_out, int out_size, void* d_ws, size_t ws_size,
                              hipStream_t stream)
{
    const float* x      = (const float*)d_in[0];
    const int*   ei     = (const int*)d_in[1];
    const int*   batch  = (const int*)d_in[2];
    const float* emb_W  = (const float*)d_in[3];
    const float* emb_b  = (const float*)d_in[4];
    const float* W1l    = (const float*)d_in[5];
    const float* b1l    = (const float*)d_in[6];
    const float* W1r    = (const float*)d_in[7];
    const float* W2l    = (const float*)d_in[8];
    const float* b2l    = (const float*)d_in[9];
    const float* W2r    = (const float*)d_in[10];
    const float* W3l    = (const float*)d_in[11];
    const float* b3l    = (const float*)d_in[12];
    const float* W3r    = (const float*)d_in[13];
    const float* flat_W = (const float*)d_in[14];
    const float* flat_b = (const float*)d_in[15];

    const int* src = ei;            // edge_index[0, :]
    const int* dst = ei + N_EDGES;  // edge_index[1, :]

    // Workspace layout (floats). h3 aliases h0 (h0 dead after layer-1 GEMM).
    float* ws = (float*)d_ws;
    size_t off = 0;
    float* h0     = ws + off; off += (size_t)N_NODES * 128;
    float* agg    = ws + off; off += (size_t)N_NODES * 128;   // reused each layer
    float* h1     = ws + off; off += (size_t)N_NODES * 32;
    float* h2     = ws + off; off += (size_t)N_NODES * 64;
    float* sums   = ws + off; off += (size_t)N_GRAPHS * 128;
    float* cnts   = ws + off; off += (size_t)N_GRAPHS;
    float* pooled = ws + off; off += (size_t)N_GRAPHS * 128;
    float* h3     = h0;

    const dim3 blk(32 * WAVES_PER_BLOCK);
    const int tpb = 256;

    // 1) h0 = x @ emb_W + emb_b                                  [50000,128]
    fused_gemm_wmma_kernel<false><<<gemm_grid(N_NODES, 128), blk, 0, stream>>>(
        x, emb_W, nullptr, nullptr, emb_b, h0, N_NODES, 128, 0, 128, 0);

    // 2) agg = segment_sum(h0[src], dst)                          d = 128
    {
        long long n = (long long)N_NODES * 128;
        zero_kernel<<<(unsigned)((n + tpb - 1) / tpb), tpb, 0, stream>>>(agg, n);
        long long tot = (long long)N_EDGES * 32;
        scatter_add_kernel<<<(unsigned)((tot + tpb - 1) / tpb), tpb, 0, stream>>>(
            h0, src, dst, agg, N_EDGES, 32, 128);
    }
    // 3) h1 = leaky(agg @ W1l + b1l + h0 @ W1r)                  [50000,32]
    fused_gemm_wmma_kernel<false><<<gemm_grid(N_NODES, 32), blk, 0, stream>>>(
        agg, W1l, h0, W1r, b1l, h1, N_NODES, 128, 128, 32, 1);

    // 4) agg = segment_sum(h1[src], dst)                          d = 32
    {
        long long n = (long long)N_NODES * 32;
        zero_kernel<<<(unsigned)((n + tpb - 1) / tpb), tpb, 0, stream>>>(agg, n);
        long long tot = (long long)N_EDGES * 8;
        scatter_add_kernel<<<(unsigned)((tot + tpb - 1) / tpb), tpb, 0, stream>>>(
            h1, src, dst, agg, N_EDGES, 8, 32);
    }
    // 5) h2 = leaky(agg @ W2l + b2l + h1 @ W2r)                  [50000,64]
    fused_gemm_wmma_kernel<false><<<gemm_grid(N_NODES, 64), blk, 0, stream>>>(
        agg, W2l, h1, W2r, b2l, h2, N_NODES, 32, 32, 64, 1);

    // 6) agg = segment_sum(h2[src], dst)                          d = 64
    {
        long long n = (long long)N_NODES * 64;
        zero_kernel<<<(unsigned)((n + tpb - 1) / tpb), tpb, 0, stream>>>(agg, n);
        long long tot = (long long)N_EDGES * 16;
        scatter_add_kernel<<<(unsigned)((tot + tpb - 1) / tpb), tpb, 0, stream>>>(
            h2, src, dst, agg, N_EDGES, 16, 64);
    }
    // 7) h3 = agg @ W3l + b3l + h2 @ W3r   (no activation)       [50000,128]
    fused_gemm_wmma_kernel<false><<<gemm_grid(N_NODES, 128), blk, 0, stream>>>(
        agg, W3l, h2, W3r, b3l, h3, N_NODES, 64, 64, 128, 0);

    // 8) global mean pool
    {
        long long n = (long long)N_GRAPHS * 128;
        zero_kernel<<<(unsigned)((n + tpb - 1) / tpb), tpb, 0, stream>>>(sums, n);
        zero_kernel<<<(unsigned)((N_GRAPHS + tpb - 1) / tpb), tpb, 0, stream>>>(cnts, N_GRAPHS);
        long long tot = (long long)N_NODES * 32;
        pool_sum_kernel<<<(unsigned)((tot + tpb - 1) / tpb), tpb, 0, stream>>>(
            h3, batch, sums, cnts, N_NODES, 32, 128);
        pool_div_kernel<<<(unsigned)((N_GRAPHS * 128 + tpb - 1) / tpb), tpb, 0, stream>>>(
            sums, cnts, pooled, N_GRAPHS, 128);
    }
    // 9) out = pooled @ flat_W + flat_b                          [1000,256]
    fused_gemm_wmma_kernel<true><<<gemm_grid(N_GRAPHS, 256), blk, 0, stream>>>(
        pooled, flat_W, nullptr, nullptr, flat_b, (float*)d_out,
        N_GRAPHS, 128, 0, 256, 0);
}